// PendulumSolver_55628416418512
// MI455X (gfx1250) — compile-verified
//
#include <hip/hip_runtime.h>
#include <math.h>

// Pendulum RK4 sweep for MI455X (gfx1250, wave32).
// - 1 thread = 1 pendulum, 999 RK4 steps, theta streamed out every step.
// - Forcing cos(2*pi*phi*t) advanced by a unit-complex rotation (no cos in loop).
// - sin(theta) via native v_sin_f32 (__sinf).
// - Output (B,1000) f32 staged per-wave in LDS (32 rows x 32 t, rows padded to
//   36 dwords) and streamed with global_store_async_from_lds_b128: 512 B per
//   wave-store (4 row segments of 128 B). Intra-wave only -> no barriers; ordering via
//   s_wait_dscnt / s_wait_asynccnt. Double-buffered so DMA overlaps compute.

#define LEN     1000
#define DTf     0.01f
#define TILE_T  32
#define ROW_PAD 36              // dwords/row: 16B-aligned rows, <=2-way LDS bank conflict
#define BLOCK   128
#define WAVES   (BLOCK / 32)

__global__ __launch_bounds__(BLOCK) void pendulum_rk4_stream(
    const float* __restrict__ init_conds,   // (B,2) [theta, thetadot]
    const float* __restrict__ params,       // (B,4) [omega, gamma, A, phi]
    float* __restrict__ out,                // (B,1000)
    int nB)
{
  __shared__ __align__(16) float tile[WAVES][2][32 * ROW_PAD];

  const int tid  = blockIdx.x * BLOCK + threadIdx.x;
  const int lane = threadIdx.x & 31;
  const int wv   = threadIdx.x >> 5;
  const int pb   = tid - lane;             // this wave's first pendulum
  if (pb >= nB) return;                    // wave-uniform exit only

  // Coalesced parameter loads: float2 / float4 per lane.
  const float2 y0 = ((const float2*)init_conds)[tid];
  const float4 p  = ((const float4*)params)[tid];
  const float omega = p.x, gamma = p.y, A = p.z, phi = p.w;

  float th = y0.x;
  float w  = y0.y;
  const float o2  = omega * omega;
  const float F   = A * o2;                 // force amplitude
  const float hdt = 0.5f * DTf;
  const float dt6 = DTf / 6.0f;

  // Half-step rotation for forcing phase: dang = 2*pi*phi*(dt/2) = pi*phi*dt.
  const float dang = 3.14159265358979f * phi * DTf;
  const float cd = __builtin_cosf(dang);    // precise, evaluated once
  const float sd = __builtin_sinf(dang);
  float cs = 1.0f, sn = 0.0f;               // cos/sin(2*pi*phi*t) at t = 0

  int buf = 0;
  tile[wv][buf][lane * ROW_PAD + 0] = th;   // column t = 0: initial condition

  for (int s = 1; s < LEN; ++s) {
    // ---- forcing values at t, t+dt/2, t+dt via two rotations ----
    const float f1 = F * cs;
    const float c1 = cs * cd - sn * sd;
    const float s1 = sn * cd + cs * sd;
    const float f2 = F * c1;                // shared by k2 and k3
    const float c2 = c1 * cd - s1 * sd;
    const float s2 = s1 * cd + c1 * sd;
    const float f4 = F * c2;
    cs = c2; sn = s2;

    // ---- classic RK4 on (theta, thetadot) ----
    const float k1t = w;
    const float k1w = f1 - gamma * w   - o2 * __sinf(th);
    const float y2t = th + hdt * k1t,  y2w = w + hdt * k1w;
    const float k2t = y2w;
    const float k2w = f2 - gamma * y2w - o2 * __sinf(y2t);
    const float y3t = th + hdt * k2t,  y3w = w + hdt * k2w;
    const float k3t = y3w;
    const float k3w = f2 - gamma * y3w - o2 * __sinf(y3t);
    const float y4t = th + DTf * k3t,  y4w = w + DTf * k3w;
    const float k4t = y4w;
    const float k4w = f4 - gamma * y4w - o2 * __sinf(y4t);
    th += dt6 * (k1t + 2.0f * (k2t + k3t) + k4t);
    w  += dt6 * (k1w + 2.0f * (k2w + k3w) + k4w);

    const int col = s & (TILE_T - 1);
    tile[wv][buf][lane * ROW_PAD + col] = th;

    if (col == TILE_T - 1 || s == LEN - 1) {
      const int t0    = s - col;
      const int width = col + 1;            // 32 for full tiles, 8 for the tail
      // LDS fills from this wave must be visible before the async engine reads.
      asm volatile("s_wait_dscnt 0" ::: "memory");

      const unsigned base = (unsigned)(size_t)&tile[wv][buf][0]; // low 32b = LDS offset
      const int rq = lane >> 3;             // row within 4-row chunk
      const int q  = lane & 7;              // position within row

      if (width == TILE_T) {
        // 8 async b128 stores: 32 lanes x 16 B = 512 B (4 rows x 128 B, rows 4000 B apart)
        #pragma unroll
        for (int c = 0; c < 8; ++c) {
          const int row = 4 * c + rq;
          const unsigned laddr = base + (unsigned)(row * ROW_PAD + q * 4) * 4u;
          const unsigned long long g =
              (unsigned long long)(out + (size_t)(pb + row) * LEN + t0 + q * 4);
          asm volatile("global_store_async_from_lds_b128 %0, %1, off"
                       :: "v"(g), "v"(laddr) : "memory");
        }
      } else {
        // tail: 8 columns -> b32 per lane (4 rows x 32 B per instruction)
        #pragma unroll
        for (int c = 0; c < 8; ++c) {
          const int row = 4 * c + rq;
          const unsigned laddr = base + (unsigned)(row * ROW_PAD + q) * 4u;
          const unsigned long long g =
              (unsigned long long)(out + (size_t)(pb + row) * LEN + t0 + q);
          asm volatile("global_store_async_from_lds_b32 %0, %1, off"
                       :: "v"(g), "v"(laddr) : "memory");
        }
      }
      buf ^= 1;
      // Double buffer: allow the 8 ops just issued in flight; anything older
      // (the previous tile's flush) must be complete before we refill its buffer.
      asm volatile("s_wait_asynccnt 8" ::: "memory");

      // Keep the forcing rotation on the unit circle (1 Newton step, every 32 steps).
      const float m = cs * cs + sn * sn;
      const float r = 0.5f * (3.0f - m);
      cs *= r; sn *= r;
    }
  }
  asm volatile("s_wait_asynccnt 0" ::: "memory");
}

extern "C" void kernel_launch(void* const* d_in, const int* in_sizes, int n_in,
                              void* d_out, int out_size, void* d_ws, size_t ws_size,
                              hipStream_t stream) {
  (void)n_in; (void)out_size; (void)d_ws; (void)ws_size;
  const float* init_conds = (const float*)d_in[0];   // (B,2) f32
  const float* params     = (const float*)d_in[1];   // (B,4) f32
  float* out = (float*)d_out;                        // (B,1000) f32
  const int nB = in_sizes[0] / 2;
  const int grid = (nB + BLOCK - 1) / BLOCK;
  pendulum_rk4_stream<<<grid, BLOCK, 0, stream>>>(init_conds, params, out, nB);
}